// AttnDecoder_50070728736834
// MI455X (gfx1250) — compile-verified
//
#include <hip/hip_runtime.h>
#include <hip/hip_bf16.h>

typedef float v2f __attribute__((ext_vector_type(2)));
typedef float v8f __attribute__((ext_vector_type(8)));

#define B_  64
#define L_  128
#define E_  512
#define H_  512
#define V_  32000
#define T_  127   // L-1 steps

// ---------------------------------------------------------------------------
// Zero-init (h0 = 0)
// ---------------------------------------------------------------------------
__global__ void zero_kernel(float* __restrict__ p, int n) {
    int i = blockIdx.x * blockDim.x + threadIdx.x;
    if (i < n) p[i] = 0.0f;
}

// ---------------------------------------------------------------------------
// Embedding gather into time-major layout: emb_t[t*B + b][e] = table[idx[b,t]][e]
// One block per (t,b) row; 128 threads x float4 = 512 floats.
// ---------------------------------------------------------------------------
__global__ void embed_kernel(const float* __restrict__ emb_table,
                             const int* __restrict__ idx,
                             float* __restrict__ emb_t) {
    int r = blockIdx.x;           // r = t*B + b
    int t = r >> 6;               // /B_
    int b = r & (B_ - 1);
    int tok = idx[b * L_ + t];
    const float4* src = (const float4*)(emb_table + (size_t)tok * E_);
    float4* dst = (float4*)(emb_t + (size_t)r * E_);
    dst[threadIdx.x] = src[threadIdx.x];
}

// ---------------------------------------------------------------------------
// Generic fp32 WMMA GEMM:  C[M,N] = A[M,K] * W[N,K]^T  (+ bias[n]) (+ addmat[m,n])
// Grid: x = N/64 (4 waves x 16 cols), y = M/16.  Block = 128 threads (4 waves).
// Uses V_WMMA_F32_16X16X4_F32; fp32 math end-to-end (matches fp32 reference).
//
// VGPR layouts per CDNA5 ISA 7.12.2 (g = lane>>4, ln = lane&15):
//   A frag (2 VGPRs): lane holds A[m=ln][k0 + 2g + v], v=0,1  -> one float2 load
//   B frag (2 VGPRs): lane holds B[k0+2g+v][n=ln] = W[n][k0+2g+v] -> one float2 load
//   C/D  (8 VGPRs):   lane/vgpr v holds C[m0 + v + 8g][n0 + ln]
// ---------------------------------------------------------------------------
__global__ void gemm_wmma_f32(const float* __restrict__ A, int lda,
                              const float* __restrict__ W, int ldw,
                              const float* __restrict__ bias,
                              const float* __restrict__ addmat, int ldadd,
                              float* __restrict__ C, int ldc, int K) {
    const int wave = threadIdx.x >> 5;
    const int lane = threadIdx.x & 31;
    const int g    = lane >> 4;
    const int ln   = lane & 15;
    const int m0   = blockIdx.y << 4;
    const int n0   = (blockIdx.x << 6) + (wave << 4);

    const float* Arow = A + (size_t)(m0 + ln) * lda + 2 * g;
    const float* Wrow = W + (size_t)(n0 + ln) * ldw + 2 * g;

    v8f acc;
    {
        float bv = bias ? bias[n0 + ln] : 0.0f;
#pragma unroll
        for (int v = 0; v < 8; ++v) {
            float av = bv;
            if (addmat)
                av += addmat[(size_t)(m0 + v + 8 * g) * ldadd + (n0 + ln)];
            acc[v] = av;
        }
    }

    for (int k = 0; k < K; k += 4) {
        v2f a = *(const v2f*)(Arow + k);
        v2f b = *(const v2f*)(Wrow + k);
        acc = __builtin_amdgcn_wmma_f32_16x16x4_f32(
            /*neg_a=*/false, a, /*neg_b=*/false, b,
            /*c_mod=*/(short)0, acc, /*reuse_a=*/false, /*reuse_b=*/false);
    }

#pragma unroll
    for (int v = 0; v < 8; ++v)
        C[(size_t)(m0 + v + 8 * g) * ldc + (n0 + ln)] = acc[v];
}

// ---------------------------------------------------------------------------
// Attention softmax: one wave per batch row of 128 scores -> d_out attns slab.
// ---------------------------------------------------------------------------
__global__ void attn_softmax_kernel(const float* __restrict__ scores,
                                    float* __restrict__ out) {
    int b = blockIdx.x;
    int l = threadIdx.x;  // 0..31
    const float* row = scores + b * L_;
    float x0 = row[l], x1 = row[l + 32], x2 = row[l + 64], x3 = row[l + 96];
    float mx = fmaxf(fmaxf(x0, x1), fmaxf(x2, x3));
#pragma unroll
    for (int o = 16; o >= 1; o >>= 1) mx = fmaxf(mx, __shfl_xor(mx, o, 32));
    float e0 = expf(x0 - mx), e1 = expf(x1 - mx);
    float e2 = expf(x2 - mx), e3 = expf(x3 - mx);
    float s = e0 + e1 + e2 + e3;
#pragma unroll
    for (int o = 16; o >= 1; o >>= 1) s += __shfl_xor(s, o, 32);
    float inv = 1.0f / s;
    float* orow = out + b * L_;
    orow[l]      = e0 * inv;
    orow[l + 32] = e1 * inv;
    orow[l + 64] = e2 * inv;
    orow[l + 96] = e3 * inv;
}

// ---------------------------------------------------------------------------
// GRU cell elementwise (PyTorch gate order r,z,n); gi/gh already include biases.
// ---------------------------------------------------------------------------
__global__ void gru_kernel(const float* __restrict__ gi,
                           const float* __restrict__ gh,
                           const float* __restrict__ h_old,
                           float* __restrict__ h_new,
                           float* __restrict__ h_fin) {
    int i = blockIdx.x * blockDim.x + threadIdx.x;  // 0..B*H
    int b = i >> 9;              // /H_
    int j = i & (H_ - 1);
    const float* gib = gi + (size_t)b * (3 * H_);
    const float* ghb = gh + (size_t)b * (3 * H_);
    float r = 1.0f / (1.0f + expf(-(gib[j] + ghb[j])));
    float z = 1.0f / (1.0f + expf(-(gib[H_ + j] + ghb[H_ + j])));
    float n = tanhf(gib[2 * H_ + j] + r * ghb[2 * H_ + j]);
    float hv = (1.0f - z) * n + z * h_old[i];
    h_new[i] = hv;
    if (h_fin) h_fin[i] = hv;
}

// ---------------------------------------------------------------------------
// Per-row CE: ce_part[t*B+b] = (logsumexp(logits[b]) - logits[b,tgt]) / B
// ---------------------------------------------------------------------------
__global__ void ce_kernel(const float* __restrict__ logits,
                          const int* __restrict__ dec_idx,
                          int t, float* __restrict__ ce_part) {
    __shared__ float red[256];
    int b = blockIdx.x;
    const float* row = logits + (size_t)b * V_;
    float mx = -3.4e38f;
    for (int i = threadIdx.x; i < V_; i += 256) mx = fmaxf(mx, row[i]);
    red[threadIdx.x] = mx; __syncthreads();
    for (int s = 128; s >= 1; s >>= 1) {
        if (threadIdx.x < s)
            red[threadIdx.x] = fmaxf(red[threadIdx.x], red[threadIdx.x + s]);
        __syncthreads();
    }
    mx = red[0]; __syncthreads();
    float sum = 0.0f;
    for (int i = threadIdx.x; i < V_; i += 256) sum += expf(row[i] - mx);
    red[threadIdx.x] = sum; __syncthreads();
    for (int s = 128; s >= 1; s >>= 1) {
        if (threadIdx.x < s) red[threadIdx.x] += red[threadIdx.x + s];
        __syncthreads();
    }
    if (threadIdx.x == 0) {
        int tgt = dec_idx[b * L_ + t];
        ce_part[t * B_ + b] = (logf(red[0]) + mx - row[tgt]) * (1.0f / B_);
    }
}

// Deterministic fixed-order final loss reduction (no float atomics).
__global__ void loss_reduce_kernel(const float* __restrict__ ce_part,
                                   float* __restrict__ loss) {
    __shared__ float red[256];
    float s = 0.0f;
    for (int i = threadIdx.x; i < T_ * B_; i += 256) s += ce_part[i];
    red[threadIdx.x] = s; __syncthreads();
    for (int st = 128; st >= 1; st >>= 1) {
        if (threadIdx.x < st) red[threadIdx.x] += red[threadIdx.x + st];
        __syncthreads();
    }
    if (threadIdx.x == 0) *loss = red[0];
}

// ---------------------------------------------------------------------------
extern "C" void kernel_launch(void* const* d_in, const int* in_sizes, int n_in,
                              void* d_out_v, int out_size, void* d_ws, size_t ws_size,
                              hipStream_t stream) {
    (void)in_sizes; (void)n_in; (void)out_size; (void)ws_size;

    const int*   dec_idx = (const int*)d_in[0];
    // d_in[1] hidden: zeroed in forward -> unused
    // d_in[2] encoder_outputs: attn_applied discarded -> unused
    const float* emb_tab = (const float*)d_in[3];
    const float* W_attn  = (const float*)d_in[4];   // [L, 2H]
    const float* b_attn  = (const float*)d_in[5];
    const float* W_ih    = (const float*)d_in[6];   // [3H, E]
    const float* W_hh    = (const float*)d_in[7];   // [3H, H]
    const float* b_ih    = (const float*)d_in[8];
    const float* b_hh    = (const float*)d_in[9];
    const float* W_out   = (const float*)d_in[10];  // [V, H]
    const float* b_out   = (const float*)d_in[11];
    float* d_out = (float*)d_out_v;

    // Workspace layout (floats), ~80 MB total
    float* ws      = (float*)d_ws;
    float* emb_t   = ws;                                  // [L*B, E]
    float* gi_all  = emb_t  + (size_t)L_ * B_ * E_;       // [T*B, 3H]
    float* dp_all  = gi_all + (size_t)T_ * B_ * 3 * H_;   // [T*B, L] (dec part + b_attn)
    float* hbuf0   = dp_all + (size_t)T_ * B_ * L_;       // [B, H]
    float* hbuf1   = hbuf0  + B_ * H_;                    // [B, H]
    float* gh_buf  = hbuf1  + B_ * H_;                    // [B, 3H]
    float* scores  = gh_buf + B_ * 3 * H_;                // [B, L]
    float* logits  = scores + B_ * L_;                    // [B, V]
    float* ce_part = logits + (size_t)B_ * V_;            // [T*B]

    // Output layout: loss(1) | attns[T,B,L] | h_fin[B,H] | logits_last[B,V]
    float* attns_out  = d_out + 1;
    float* hfin_out   = attns_out + (size_t)T_ * B_ * L_;
    float* logits_out = hfin_out  + (size_t)B_ * H_;

    // h0 = 0; embeddings (time-major so batched GEMM rows are contiguous)
    zero_kernel<<<(B_ * H_ + 255) / 256, 256, 0, stream>>>(hbuf0, B_ * H_);
    embed_kernel<<<L_ * B_, 128, 0, stream>>>(emb_tab, dec_idx, emb_t);

    // Hoisted batched GEMMs over all 127 steps (h-independent):
    //   gi_all[t*B+b] = emb_t[t*B+b] @ W_ih^T + b_ih
    gemm_wmma_f32<<<dim3(3 * H_ / 64, T_ * B_ / 16), 128, 0, stream>>>(
        emb_t, E_, W_ih, E_, b_ih, nullptr, 0, gi_all, 3 * H_, E_);
    //   dp_all[t*B+b] = emb_t[t*B+b] @ W_attn[:, :H]^T + b_attn
    gemm_wmma_f32<<<dim3(L_ / 64, T_ * B_ / 16), 128, 0, stream>>>(
        emb_t, E_, W_attn, 2 * H_, b_attn, nullptr, 0, dp_all, L_, E_);

    float* h_old = hbuf0;
    float* h_new = hbuf1;
    for (int t = 0; t < T_; ++t) {
        int prev = t > 0 ? t - 1 : 0;  // dec_in at step t is embedded[:, max(t-1,0)]

        // scores = dp_all[prev block] + h_old @ W_attn[:, H:]^T
        gemm_wmma_f32<<<dim3(L_ / 64, B_ / 16), 128, 0, stream>>>(
            h_old, H_, W_attn + H_, 2 * H_, nullptr,
            dp_all + (size_t)prev * B_ * L_, L_, scores, L_, H_);
        attn_softmax_kernel<<<B_, 32, 0, stream>>>(
            scores, attns_out + (size_t)t * B_ * L_);

        // gh = h_old @ W_hh^T + b_hh
        gemm_wmma_f32<<<dim3(3 * H_ / 64, B_ / 16), 128, 0, stream>>>(
            h_old, H_, W_hh, H_, b_hh, nullptr, 0, gh_buf, 3 * H_, H_);

        gru_kernel<<<(B_ * H_) / 256, 256, 0, stream>>>(
            gi_all + (size_t)t * B_ * 3 * H_, gh_buf, h_old, h_new,
            (t == T_ - 1) ? hfin_out : nullptr);

        // logits = h_new @ W_out^T + b_out   (W_out stays L2-resident: 65.5MB < 192MB)
        float* lptr = (t == T_ - 1) ? logits_out : logits;
        gemm_wmma_f32<<<dim3(V_ / 64, B_ / 16), 128, 0, stream>>>(
            h_new, H_, W_out, H_, b_out, nullptr, 0, lptr, V_, H_);

        ce_kernel<<<B_, 256, 0, stream>>>(lptr, dec_idx, t, ce_part);

        float* tmp = h_old; h_old = h_new; h_new = tmp;
    }

    loss_reduce_kernel<<<1, 256, 0, stream>>>(ce_part, d_out);
}